// BfpQuantizer_80144089743883
// MI455X (gfx1250) — compile-verified
//
#include <hip/hip_runtime.h>
#include <stdint.h>

// ---------------------------------------------------------------------------
// BFP block quantizer for MI455X (gfx1250).
// Memory-bound streaming kernel: 512 MB traffic @ 23.3 TB/s ~= 22 us floor.
// Uses the CDNA5 Tensor Data Mover (tensor_load_to_lds) to stage 16KB tiles
// into LDS with double buffering (TENSORcnt-paced), computes with wave32
// lane-pair reductions, and writes back with non-temporal 128-bit stores.
// ---------------------------------------------------------------------------

typedef float    f32x4 __attribute__((ext_vector_type(4)));
typedef unsigned u32x4 __attribute__((ext_vector_type(4)));
typedef int      i32x4 __attribute__((ext_vector_type(4)));
typedef int      i32x8 __attribute__((ext_vector_type(8)));

#define BFPQ_THREADS      256
#define BFPQ_TILE_FLOATS  4096                              // 16 KB tile
#define BFPQ_F4_PER_THR   (BFPQ_TILE_FLOATS / (BFPQ_THREADS * 4))  // = 4
#define BFPQ_MAX_GRID     2048u

#if __has_builtin(__builtin_amdgcn_tensor_load_to_lds)
#define BFPQ_HAVE_TDM 1
#else
#define BFPQ_HAVE_TDM 0
#endif

// bf16 round-to-nearest-even of an fp32 bit pattern, saturating to
// +-(2 - 2^-7)*2^127 instead of infinity (matches float_quantize(8,7)).
__device__ __forceinline__ unsigned bf16_rne_sat_bits(unsigned u) {
  unsigned r = (u + 0x7FFFu + ((u >> 16) & 1u)) & 0xFFFF0000u;
  if ((r & 0x7F800000u) == 0x7F800000u)            // overflowed to inf
    r = (r & 0x80000000u) | 0x7F7F0000u;           // clamp to max bf16
  return r;
}

#if BFPQ_HAVE_TDM
// Issue one 1-D TDM tile load: global (valid_elems fp32, zero-fill OOB)
// -> LDS at lds_byte_addr, tile_dim0 = BFPQ_TILE_FLOATS elements.
// D# packing per CDNA5 ISA 8.3-8.6 (group0 128b, group1 256b).
__device__ __forceinline__ void tdm_load_tile_1d(const float* gsrc,
                                                 unsigned lds_byte_addr,
                                                 unsigned valid_elems) {
  uint64_t ga = (uint64_t)(uintptr_t)gsrc;
  u32x4 g0;
  g0[0] = 1u;                                       // count=1 (valid), user mode
  g0[1] = lds_byte_addr;                            // lds_addr [63:32]
  g0[2] = (unsigned)ga;                             // global_addr [95:64]
  g0[3] = (unsigned)((ga >> 32) & 0x01FFFFFFull)    // global_addr [120:96]
        | (2u << 30);                               // type=2 ("image")

  i32x8 g1;
  g1[0] = (int)(2u << 16);                          // data_size=2 (4B); wg_mask=0
  g1[1] = (int)((valid_elems & 0xFFFFu) << 16);     // tensor_dim0[15:0] @ bits 63:48
  g1[2] = (int)(((valid_elems >> 16) & 0xFFFFu)     // tensor_dim0[31:16] @ bits 79:64
        | (1u << 16));                              // tensor_dim1 = 1  @ bits 95:80
  g1[3] = (int)(((unsigned)BFPQ_TILE_FLOATS) << 16);// tile_dim0 @ bits 127:112
  g1[4] = 0;                                        // tile_dim1/2 = 0 (unused)
  g1[5] = 0; g1[6] = 0; g1[7] = 0;                  // strides unused for 1-D

  i32x4 g2 = {0, 0, 0, 0};
  i32x4 g3 = {0, 0, 0, 0};
#if __clang_major__ >= 23
  i32x8 g4 = {0, 0, 0, 0, 0, 0, 0, 0};
  __builtin_amdgcn_tensor_load_to_lds(g0, g1, g2, g3, g4, 0);
#else
  __builtin_amdgcn_tensor_load_to_lds(g0, g1, g2, g3, 0);
#endif
}
#endif  // BFPQ_HAVE_TDM

__global__ __launch_bounds__(BFPQ_THREADS)
void bfp_block_quant_kernel(const float* __restrict__ in,
                            float* __restrict__ out,
                            unsigned n) {
  __shared__ f32x4 smem[2 * BFPQ_TILE_FLOATS / 4];  // 32 KB, double buffered
  const unsigned tid     = threadIdx.x;
  const unsigned ntiles  = (n + BFPQ_TILE_FLOATS - 1) / BFPQ_TILE_FLOATS;
  const unsigned tile0   = blockIdx.x;
  const unsigned tstride = gridDim.x;
  if (tile0 >= ntiles) return;

#if BFPQ_HAVE_TDM
  const unsigned lds_base   = (unsigned)(uintptr_t)&smem[0]; // LDS byte offset
  const unsigned tile_bytes = BFPQ_TILE_FLOATS * 4u;
  if (tid < 32)                                     // wave 0 drives the TDM
    tdm_load_tile_1d(in + (size_t)tile0 * BFPQ_TILE_FLOATS, lds_base,
                     n - tile0 * BFPQ_TILE_FLOATS);
#endif

  unsigned buf = 0;
  for (unsigned t = tile0; t < ntiles; t += tstride) {
    const unsigned nt = t + tstride;
#if BFPQ_HAVE_TDM
    if (tid < 32) {
      if (nt < ntiles) {
        tdm_load_tile_1d(in + (size_t)nt * BFPQ_TILE_FLOATS,
                         lds_base + (buf ^ 1u) * tile_bytes,
                         n - nt * BFPQ_TILE_FLOATS);
        __builtin_amdgcn_s_wait_tensorcnt(1);       // tile t landed, t+1 in flight
      } else {
        __builtin_amdgcn_s_wait_tensorcnt(0);
      }
    }
    __syncthreads();                                 // publish LDS tile t to all waves
    const f32x4* __restrict__ src4 = &smem[(buf * BFPQ_TILE_FLOATS) / 4];
#else
    const f32x4* __restrict__ src4 =
        (const f32x4*)(in + (size_t)t * BFPQ_TILE_FLOATS);
#endif

    const size_t gelem0 = (size_t)t * BFPQ_TILE_FLOATS;
#pragma unroll
    for (int j = 0; j < BFPQ_F4_PER_THR; ++j) {
      const unsigned idx = (unsigned)j * BFPQ_THREADS + tid;  // float4 index
#if BFPQ_HAVE_TDM
      f32x4 v = src4[idx];                           // ds_load_b128, conflict-free
#else
      f32x4 v = __builtin_nontemporal_load(&src4[idx]);
#endif
      // Stage 1: bf16 RNE quantize (saturating) each element.
      const unsigned r0 = bf16_rne_sat_bits(__float_as_uint(v.x));
      const unsigned r1 = bf16_rne_sat_bits(__float_as_uint(v.y));
      const unsigned r2 = bf16_rne_sat_bits(__float_as_uint(v.z));
      const unsigned r3 = bf16_rne_sat_bits(__float_as_uint(v.w));

      // Stage 2: block max over the 8-float block = this lane's 4 + partner's 4.
      // Positive-float compare == unsigned int compare on |bits|.
      const unsigned a0 = r0 & 0x7FFFFFFFu, a1 = r1 & 0x7FFFFFFFu;
      const unsigned a2 = r2 & 0x7FFFFFFFu, a3 = r3 & 0x7FFFFFFFu;
      unsigned m01 = a0 > a1 ? a0 : a1;
      unsigned m23 = a2 > a3 ? a2 : a3;
      unsigned mb  = m01 > m23 ? m01 : m23;
      float mf = __uint_as_float(mb);
      mf = fmaxf(mf, __shfl_xor(mf, 1, 32));         // lane pair (2i, 2i+1)
      mb = __float_as_uint(mf);

      // exponent = floor(log2(max)); shared exponent scale = 2^(E-6).
      int E = (int)(mb >> 23) - 127;
      if (E < -120) E = -120;                        // keep scales normal; m==0 -> q==0
      const float inv_s = __uint_as_float((unsigned)(133 - E) << 23); // 2^(6-E)
      const float s     = __uint_as_float((unsigned)(E + 121) << 23); // 2^(E-6)

      f32x4 q;
      q.x = fminf(fmaxf(rintf(__uint_as_float(r0) * inv_s), -127.0f), 127.0f) * s;
      q.y = fminf(fmaxf(rintf(__uint_as_float(r1) * inv_s), -127.0f), 127.0f) * s;
      q.z = fminf(fmaxf(rintf(__uint_as_float(r2) * inv_s), -127.0f), 127.0f) * s;
      q.w = fminf(fmaxf(rintf(__uint_as_float(r3) * inv_s), -127.0f), 127.0f) * s;

      const size_t gi = gelem0 + (size_t)idx * 4u;
      if (gi + 4u <= (size_t)n) {
        __builtin_nontemporal_store(q, (f32x4*)(out + gi));  // streaming store
      } else {
        const float qa[4] = {q.x, q.y, q.z, q.w};
        for (int k = 0; k < 4; ++k)
          if (gi + (unsigned)k < (size_t)n) out[gi + (unsigned)k] = qa[k];
      }
    }
#if BFPQ_HAVE_TDM
    __syncthreads();                                 // all done reading buf before reuse
#endif
    buf ^= 1u;
  }
}

extern "C" void kernel_launch(void* const* d_in, const int* in_sizes, int n_in,
                              void* d_out, int out_size, void* d_ws, size_t ws_size,
                              hipStream_t stream) {
  (void)n_in; (void)d_ws; (void)ws_size; (void)in_sizes;
  const float* x = (const float*)d_in[0];
  float* y = (float*)d_out;
  const unsigned n = (unsigned)out_size;             // 8*2048*4096 = 2^26
  if (n == 0) return;
  const unsigned ntiles = (n + BFPQ_TILE_FLOATS - 1) / BFPQ_TILE_FLOATS;
  const unsigned grid = ntiles < BFPQ_MAX_GRID ? ntiles : BFPQ_MAX_GRID;
  bfp_block_quant_kernel<<<dim3(grid), dim3(BFPQ_THREADS), 0, stream>>>(x, y, n);
}